// Feature2VertexLayer_10342281249418
// MI455X (gfx1250) — compile-verified
//
#include <hip/hip_runtime.h>

// ---------------------------------------------------------------------------
// GraphConvNorm stack for MI455X (gfx1250), fp32 end-to-end.
// GEMMs use V_WMMA_F32_16X16X4_F32 (wave32 matrix pipe, exact fp32 numerics).
// Scatter-add uses global_atomic_add_f32 (L2-resident working set).
// ---------------------------------------------------------------------------

#define N_VERT 100000
#define N_EDGE 1600000

typedef float v2f __attribute__((ext_vector_type(2)));
typedef float v8f __attribute__((ext_vector_type(8)));

// ---------------------------------------------------------------- utilities
__global__ __launch_bounds__(256) void zero_f32(float* __restrict__ p, int n) {
  int i = blockIdx.x * blockDim.x + threadIdx.x;
  if (i < n) p[i] = 0.0f;
}

// degree[v] += 1 for every occurrence of v in the flattened edge list (2E ints)
__global__ __launch_bounds__(256) void degree_kernel(const int* __restrict__ ev,
                                                     float* __restrict__ deg, int n) {
  int i = blockIdx.x * blockDim.x + threadIdx.x;
  if (i < n) atomicAdd(deg + ev[i], 1.0f);
}

// ------------------------------------------------------- dual GEMM via WMMA
// One wave computes a 16x16 output tile of BOTH  H0 = X*W0 + b0  and
// H1 = X*W1 + b1, sharing the A (X) tile between the two accumulators.
//
// V_WMMA_F32_16X16X4_F32 operand layouts (ISA 7.12.2):
//   A (16x4):  lane 0..15 -> M=lane, VGPR0/1 = K0/K1 ; lane 16..31 -> K2/K3
//   B (4x16):  lane 0..15 -> N=lane, VGPR0/1 = K0/K1 ; lane 16..31 -> K2/K3
//   C (16x16): VGPR j: lanes 0..15 -> M=j, lanes 16..31 -> M=j+8, N=lane&15
template <int DIN, int DOUT>
__global__ __launch_bounds__(128)
void gemm_dual_wmma(const float* __restrict__ X,
                    const float* __restrict__ W0, const float* __restrict__ b0,
                    const float* __restrict__ W1, const float* __restrict__ b1,
                    float* __restrict__ H0, float* __restrict__ H1) {
  const int lane = threadIdx.x & 31;
  const int wave = threadIdx.x >> 5;

  const int mt = blockIdx.x * 4 + wave;           // M tile index (wave-uniform)
  if (mt * 16 >= N_VERT) return;                  // uniform per wave: EXEC stays full
  const int m0 = mt * 16;
  const int n0 = blockIdx.y * 16;

  const int mrow  = lane & 15;                    // A row this lane holds
  const int ncol  = lane & 15;                    // B/C column this lane holds
  const int khalf = (lane >> 4) << 1;             // 0 for lanes 0-15, 2 for 16-31

  // column handling for the ragged last layer (DOUT == 3)
  int   nc;   // clamped column for safe addressing
  float msk;  // 0/1 validity multiplier
  if constexpr (DOUT % 16 == 0) {
    nc = n0 + ncol; msk = 1.0f;
  } else {
    const int ng = n0 + ncol;
    nc  = (ng < DOUT) ? ng : 0;
    msk = (ng < DOUT) ? 1.0f : 0.0f;
  }

  v8f c0 = {};
  v8f c1 = {};
  const float* __restrict__ xrow = X + (size_t)(m0 + mrow) * DIN;

#pragma unroll 4
  for (int k = 0; k < DIN; k += 4) {
    v2f a;
    a.x = xrow[k + khalf];                        // merges to one b64 load
    a.y = xrow[k + khalf + 1];

    v2f bA, bB;
    bA.x = W0[(size_t)(k + khalf)     * DOUT + nc] * msk;
    bA.y = W0[(size_t)(k + khalf + 1) * DOUT + nc] * msk;
    bB.x = W1[(size_t)(k + khalf)     * DOUT + nc] * msk;
    bB.y = W1[(size_t)(k + khalf + 1) * DOUT + nc] * msk;

    c0 = __builtin_amdgcn_wmma_f32_16x16x4_f32(false, a, false, bA,
                                               (short)0, c0, false, false);
    c1 = __builtin_amdgcn_wmma_f32_16x16x4_f32(false, a, false, bB,
                                               (short)0, c1, false, false);
  }

  const bool colok = (DOUT % 16 == 0) ? true : (n0 + ncol < DOUT);
  if (colok) {
    const float bias0 = b0[n0 + ncol];
    const float bias1 = b1[n0 + ncol];
    const int rbase = m0 + ((lane >> 4) << 3);    // +0 or +8
#pragma unroll
    for (int j = 0; j < 8; ++j) {
      const size_t o = (size_t)(rbase + j) * DOUT + (n0 + ncol);
      H0[o] = c0[j] + bias0;
      H1[o] = c1[j] + bias1;
    }
  }
}

// ------------------------------------------------------------- edge scatter
// AGG[src] += H1[dst]; AGG[dst] += H1[src]   (both directions per thread)
// Each thread owns CPT consecutive channels of one edge.
template <int DOUT, int CPT>
__global__ __launch_bounds__(256)
void scatter_edges(const int* __restrict__ ev, const float* __restrict__ H1,
                   float* __restrict__ AGG) {
  constexpr int GROUPS = DOUT / CPT;
  const long long idx = (long long)blockIdx.x * blockDim.x + threadIdx.x;
  if (idx >= (long long)N_EDGE * GROUPS) return;
  const int e = (int)(idx / GROUPS);
  const int c = (int)(idx % GROUPS) * CPT;

  const int vs = ev[2 * e + 0];
  const int vd = ev[2 * e + 1];

  const float* __restrict__ ps = H1 + (size_t)vs * DOUT + c;
  const float* __restrict__ pd = H1 + (size_t)vd * DOUT + c;
  float* as = AGG + (size_t)vs * DOUT + c;
  float* ad = AGG + (size_t)vd * DOUT + c;

  float rs[CPT], rd[CPT];
#pragma unroll
  for (int j = 0; j < CPT; ++j) { rs[j] = ps[j]; rd[j] = pd[j]; }
#pragma unroll
  for (int j = 0; j < CPT; ++j) {
    atomicAdd(as + j, rd[j]);
    atomicAdd(ad + j, rs[j]);
  }
}

// ----------------------------------------------------- normalize (+ReLU)
template <int DOUT, bool RELU>
__global__ __launch_bounds__(256)
void finalize_kernel(const float* __restrict__ AGG, const float* __restrict__ deg,
                     float* __restrict__ Y) {
  const long long idx = (long long)blockIdx.x * blockDim.x + threadIdx.x;
  if (idx >= (long long)N_VERT * DOUT) return;
  const int v = (int)(idx / DOUT);
  float val = AGG[idx] * (1.0f / deg[v]);
  if (RELU) val = fmaxf(val, 0.0f);
  Y[idx] = val;
}

// --------------------------------------------------------------- launcher
extern "C" void kernel_launch(void* const* d_in, const int* in_sizes, int n_in,
                              void* d_out, int out_size, void* d_ws, size_t ws_size,
                              hipStream_t stream) {
  (void)in_sizes; (void)n_in; (void)out_size; (void)ws_size;

  const float* feat  = (const float*)d_in[0];
  const int*   edges = (const int*)d_in[1];
  const float *W0[4], *B0[4], *W1[4], *B1[4];
  for (int l = 0; l < 4; ++l) {
    W0[l] = (const float*)d_in[2 + 4 * l + 0];
    B0[l] = (const float*)d_in[2 + 4 * l + 1];
    W1[l] = (const float*)d_in[2 + 4 * l + 2];
    B1[l] = (const float*)d_in[2 + 4 * l + 3];
  }
  float* outp = (float*)d_out;

  // carve workspace (256B aligned): deg + two ping-pong activation buffers + H1
  size_t off = 0;
  auto carve = [&](size_t elems) {
    float* p = (float*)((char*)d_ws + off);
    off += ((elems * sizeof(float) + 255) & ~(size_t)255);
    return p;
  };
  float* deg  = carve(N_VERT);
  float* bufA = carve((size_t)N_VERT * 192);
  float* bufB = carve((size_t)N_VERT * 192);
  float* bufH = carve((size_t)N_VERT * 192);

  // degrees (recomputed every call: deterministic, no cross-call state)
  zero_f32<<<(N_VERT + 255) / 256, 256, 0, stream>>>(deg, N_VERT);
  degree_kernel<<<(2 * N_EDGE + 255) / 256, 256, 0, stream>>>(edges, deg, 2 * N_EDGE);

  const int MT = N_VERT / 16;               // 6250 M tiles (N_VERT % 16 == 0)
  const unsigned gx = (MT + 3) / 4;         // 4 waves per block, one M tile each

  auto blocks = [](long long total) { return (unsigned)((total + 255) / 256); };

  // ---- layer 0: 256 -> 192, X = features, AGG = bufA
  gemm_dual_wmma<256, 192><<<dim3(gx, 192 / 16), 128, 0, stream>>>(
      feat, W0[0], B0[0], W1[0], B1[0], bufA, bufH);
  scatter_edges<192, 4><<<blocks((long long)N_EDGE * (192 / 4)), 256, 0, stream>>>(
      edges, bufH, bufA);
  finalize_kernel<192, true><<<blocks((long long)N_VERT * 192), 256, 0, stream>>>(
      bufA, deg, bufA);

  // ---- layer 1: 192 -> 128, X = bufA, AGG = bufB
  gemm_dual_wmma<192, 128><<<dim3(gx, 128 / 16), 128, 0, stream>>>(
      bufA, W0[1], B0[1], W1[1], B1[1], bufB, bufH);
  scatter_edges<128, 4><<<blocks((long long)N_EDGE * (128 / 4)), 256, 0, stream>>>(
      edges, bufH, bufB);
  finalize_kernel<128, true><<<blocks((long long)N_VERT * 128), 256, 0, stream>>>(
      bufB, deg, bufB);

  // ---- layer 2: 128 -> 64, X = bufB, AGG = bufA
  gemm_dual_wmma<128, 64><<<dim3(gx, 64 / 16), 128, 0, stream>>>(
      bufB, W0[2], B0[2], W1[2], B1[2], bufA, bufH);
  scatter_edges<64, 4><<<blocks((long long)N_EDGE * (64 / 4)), 256, 0, stream>>>(
      edges, bufH, bufA);
  finalize_kernel<64, true><<<blocks((long long)N_VERT * 64), 256, 0, stream>>>(
      bufA, deg, bufA);

  // ---- layer 3: 64 -> 3, X = bufA, AGG = bufB, output -> d_out (no ReLU)
  gemm_dual_wmma<64, 3><<<dim3(gx, 1), 128, 0, stream>>>(
      bufA, W0[3], B0[3], W1[3], B1[3], bufB, bufH);
  scatter_edges<3, 3><<<blocks((long long)N_EDGE * 1), 256, 0, stream>>>(
      edges, bufH, bufB);
  finalize_kernel<3, false><<<blocks((long long)N_VERT * 3), 256, 0, stream>>>(
      bufB, deg, outp);
}